// Police_56702158241989
// MI455X (gfx1250) — compile-verified
//
#include <hip/hip_runtime.h>
#include <cfloat>

// ---------------------------------------------------------------------------
// 2-layer GATv2 (heads=1, share_weights, self-loops w/ mean edge attr) +
// exact JAX threefry categorical sampling + log_prob.   MI455X / gfx1250.
// fp32 throughout (argmax-sensitive output); node GEMMs use V_WMMA_F32_16X16X4_F32.
// ---------------------------------------------------------------------------

typedef __attribute__((ext_vector_type(2))) float v2f;
typedef __attribute__((ext_vector_type(8))) float v8f;

#define D_EDGE 16
#define SLOPE  0.2f

// ---------------- atomics -------------------------------------------------

__device__ __forceinline__ void atomicAddF(float* p, float v) {
  // native global_atomic_add_f32 (round-to-nearest-even, no denorm flush)
  unsafeAtomicAdd(p, v);
}

__device__ __forceinline__ void atomicMaxF(float* addr, float value) {
  // monotonic int-order trick; works for mixed signs, init must be consistent
  if (value >= 0.0f)
    atomicMax((int*)addr, __float_as_int(value));
  else
    atomicMin((unsigned int*)addr, (unsigned int)__float_as_int(value));
}

// ---------------- utility kernels ----------------------------------------

__global__ void fillk(float* __restrict__ p, float v, int n) {
  int i = blockIdx.x * blockDim.x + threadIdx.x;
  if (i < n) p[i] = v;
}

// deg[dst] += 1 ; ea_sum[dst][k] += ea[e][k]      (one thread per (e,k))
__global__ void deg_easum_kernel(const int* __restrict__ dst,
                                 const float* __restrict__ ea,
                                 float* __restrict__ deg,
                                 float* __restrict__ easum, int nE) {
  int t = blockIdx.x * blockDim.x + threadIdx.x;
  if (t >= nE * D_EDGE) return;
  int e = t >> 4, k = t & 15;
  int d = dst[e];
  atomicAddF(&easum[d * D_EDGE + k], ea[t]);
  if (k == 0) atomicAddF(&deg[d], 1.0f);
}

__global__ void eamean_kernel(const float* __restrict__ deg,
                              float* __restrict__ ea_sum_to_mean, int nN) {
  int t = blockIdx.x * blockDim.x + threadIdx.x;
  if (t >= nN * D_EDGE) return;
  int node = t >> 4;
  ea_sum_to_mean[t] /= fmaxf(deg[node], 1.0f);
}

// ---------------- WMMA GEMM: out[n_rows x n_cols] = A[n_rows x 64] @ W + bias
// One wave computes one 16x16 tile; K=64 -> 16 chained v_wmma_f32_16x16x4_f32.
// A (16x4 f32) layout : lanes 0-15 row=lane, K={k0,k0+1}; lanes 16-31 K={k0+2,k0+3}
// B (4x16  f32) layout: lanes 0-15 col=lane, rows {k0,k0+1}; lanes 16-31 rows {k0+2,k0+3}
// C/D (16x16 f32)     : VGPR j -> M=j (lanes 0-15) / M=j+8 (lanes 16-31), N=lane&15
// ---------------------------------------------------------------------------
__global__ void gemm_wmma_f32(const float* __restrict__ A,
                              const float* __restrict__ W,
                              const float* __restrict__ bias,
                              float* __restrict__ out,
                              int n_rows, int n_cols) {
  const int K = 64;
  const int lane = threadIdx.x & 31;
  const int wave = blockIdx.x * (blockDim.x >> 5) + (threadIdx.x >> 5);
  const int tiles_n = n_cols >> 4;
  const int tiles_m = n_rows >> 4;
  if (wave >= tiles_m * tiles_n) return;
  const int tm = wave / tiles_n, tn = wave % tiles_n;
  const int half = lane >> 4;      // 0: lanes 0-15, 1: lanes 16-31
  const int l16  = lane & 15;
  const int arow = tm * 16 + l16;  // A row for this lane
  const int bcol = tn * 16 + l16;  // B/D column for this lane

  v8f c = {};
#pragma unroll
  for (int k0 = 0; k0 < K; k0 += 4) {
    const int ka = k0 + (half << 1);
    v2f a, b;
    a.x = A[arow * K + ka];
    a.y = A[arow * K + ka + 1];
    b.x = W[ka * n_cols + bcol];
    b.y = W[(ka + 1) * n_cols + bcol];
    c = __builtin_amdgcn_wmma_f32_16x16x4_f32(
        /*neg_a=*/false, a, /*neg_b=*/false, b,
        /*c_mod=*/(short)0, c, /*reuse_a=*/false, /*reuse_b=*/false);
  }
  const float bb = bias[bcol];
#pragma unroll
  for (int j = 0; j < 8; ++j) {
    int row = tm * 16 + j + (half << 3);
    out[row * n_cols + bcol] = c[j] + bb;
  }
}

// ---------------- edge pass 1: alpha_e = att . leakyrelu(h[s]+h[d]+ea@We) --
// one wave per edge (E real edges then N self-loops using ea_mean)
__global__ void edge_alpha_kernel(const float* __restrict__ h,
                                  const int* __restrict__ src,
                                  const int* __restrict__ dst,
                                  const float* __restrict__ ea,
                                  const float* __restrict__ ea_mean,
                                  const float* __restrict__ We,   // [16 x D]
                                  const float* __restrict__ att,  // [D]
                                  float* __restrict__ alpha,
                                  float* __restrict__ mbuf,
                                  int nE, int nN, int D) {
  __shared__ float sWe[D_EDGE * 64];
  __shared__ float sAtt[64];
  for (int i = threadIdx.x; i < D_EDGE * D; i += blockDim.x) sWe[i] = We[i];
  for (int i = threadIdx.x; i < D; i += blockDim.x) sAtt[i] = att[i];
  __syncthreads();

  const int lane = threadIdx.x & 31;
  const int e = blockIdx.x * (blockDim.x >> 5) + (threadIdx.x >> 5);
  if (e >= nE + nN) return;

  int s, d;
  const float* eap;
  if (e < nE) { s = src[e]; d = dst[e]; eap = ea + (size_t)e * D_EDGE; }
  else        { s = e - nE; d = s;      eap = ea_mean + (size_t)s * D_EDGE; }

  const float eav = eap[lane & 15];   // ea[k] in lane k (dup in upper half)
  float partial = 0.0f;
  for (int dd = lane; dd < D; dd += 32) {
    float acc = h[(size_t)s * D + dd] + h[(size_t)d * D + dd];
#pragma unroll
    for (int k = 0; k < D_EDGE; ++k)
      acc = fmaf(__shfl(eav, k, 32), sWe[k * D + dd], acc);
    float z = acc > 0.0f ? acc : SLOPE * acc;
    partial = fmaf(z, sAtt[dd], partial);
  }
#pragma unroll
  for (int off = 16; off; off >>= 1) partial += __shfl_xor(partial, off, 32);
  if (lane == 0) {
    alpha[e] = partial;
    atomicMaxF(&mbuf[d], partial);
  }
}

// ---------------- edge pass 2: ex=exp(a-m[d]); denom[d]+=ex; acc[d]+=ex*h[s]
__global__ void edge_aggr_kernel(const float* __restrict__ h,
                                 const int* __restrict__ src,
                                 const int* __restrict__ dst,
                                 const float* __restrict__ alpha,
                                 const float* __restrict__ mbuf,
                                 float* __restrict__ denom,
                                 float* __restrict__ acc,
                                 int nE, int nN, int D) {
  const int lane = threadIdx.x & 31;
  const int e = blockIdx.x * (blockDim.x >> 5) + (threadIdx.x >> 5);
  if (e >= nE + nN) return;
  int s, d;
  if (e < nE) { s = src[e]; d = dst[e]; } else { s = e - nE; d = s; }
  const float ex = expf(alpha[e] - mbuf[d]);
  if (lane == 0) atomicAddF(&denom[d], ex);
  for (int dd = lane; dd < D; dd += 32)
    atomicAddF(&acc[(size_t)d * D + dd], ex * h[(size_t)s * D + dd]);
}

// ---------------- out = acc/denom + bias ----------------------------------
__global__ void finalize_kernel(const float* __restrict__ acc,
                                const float* __restrict__ denom,
                                const float* __restrict__ bias,
                                float* __restrict__ out, int nN, int D) {
  int i = blockIdx.x * blockDim.x + threadIdx.x;
  if (i >= nN * D) return;
  int node = i / D, dd = i - node * D;
  out[i] = acc[i] / denom[node] + bias[dd];
}

// ---------------- JAX threefry2x32 (5 double-round groups) ----------------
__device__ __forceinline__ unsigned rotl32(unsigned v, int d) {
  return (v << d) | (v >> (32 - d));
}
__device__ void threefry2x32(unsigned k0, unsigned k1, unsigned c0, unsigned c1,
                             unsigned& y0, unsigned& y1) {
  unsigned ks0 = k0, ks1 = k1, ks2 = k0 ^ k1 ^ 0x1BD11BDAu;
  unsigned x0 = c0 + ks0, x1 = c1 + ks1;
#define TF_RND(r) { x0 += x1; x1 = rotl32(x1, r); x1 ^= x0; }
  TF_RND(13) TF_RND(15) TF_RND(26) TF_RND(6)
  x0 += ks1; x1 += ks2 + 1u;
  TF_RND(17) TF_RND(29) TF_RND(16) TF_RND(24)
  x0 += ks2; x1 += ks0 + 2u;
  TF_RND(13) TF_RND(15) TF_RND(26) TF_RND(6)
  x0 += ks0; x1 += ks1 + 3u;
  TF_RND(17) TF_RND(29) TF_RND(16) TF_RND(24)
  x0 += ks1; x1 += ks2 + 4u;
  TF_RND(13) TF_RND(15) TF_RND(26) TF_RND(6)
  x0 += ks2; x1 += ks0 + 5u;
#undef TF_RND
  y0 = x0; y1 = x1;
}

// ---------------- categorical sample (Gumbel argmax) + log_prob -----------
// one wave per node, lane = action; JAX key(42) -> (0, 42)
__global__ void sample_kernel(const float* __restrict__ logits,
                              float* __restrict__ out_actions,
                              float* __restrict__ out_logp, int nN) {
  const int lane = threadIdx.x & 31;
  const int node = blockIdx.x * (blockDim.x >> 5) + (threadIdx.x >> 5);
  if (node >= nN) return;

  const float logit = logits[node * 32 + lane];

  const unsigned idx  = (unsigned)(node * 32 + lane);
  const unsigned half = (unsigned)(nN * 16);       // (N*32)/2
  const unsigned p    = idx < half ? idx : idx - half;
  unsigned y0, y1;
  threefry2x32(0u, 42u, p, p + half, y0, y1);
  const unsigned bits = idx < half ? y0 : y1;

  const float tiny = 1.1754943508222875e-38f;      // FLT_MIN
  float f = __uint_as_float((bits >> 9) | 0x3F800000u) - 1.0f;   // [0,1)
  float u = fmaxf(f * (1.0f - tiny) + tiny, tiny);
  float g = -logf(-logf(u));                       // Gumbel(0,1)
  float val = logit + g;

  // argmax over lanes, first index wins ties
  int best = lane; float bv = val;
#pragma unroll
  for (int off = 16; off; off >>= 1) {
    float ov = __shfl_xor(bv, off, 32);
    int   oi = __shfl_xor(best, off, 32);
    if (ov > bv || (ov == bv && oi < best)) { bv = ov; best = oi; }
  }
  // stable logsumexp
  float ml = logit;
#pragma unroll
  for (int off = 16; off; off >>= 1) ml = fmaxf(ml, __shfl_xor(ml, off, 32));
  float ex = expf(logit - ml);
#pragma unroll
  for (int off = 16; off; off >>= 1) ex += __shfl_xor(ex, off, 32);
  const float logZ = ml + logf(ex);
  const float chosen = __shfl(logit, best, 32);

  if (lane == 0) {
    out_actions[node] = (float)best;
    out_logp[node]    = chosen - logZ;
  }
}

// ---------------------------------------------------------------------------

static inline int cdiv_i(long a, long b) { return (int)((a + b - 1) / b); }

extern "C" void kernel_launch(void* const* d_in, const int* in_sizes, int n_in,
                              void* d_out, int out_size, void* d_ws, size_t ws_size,
                              hipStream_t stream) {
  const int N = in_sizes[0] / 64;   // 50000
  const int E = in_sizes[1] / 2;    // 800000

  const float* x    = (const float*)d_in[0];
  const int*   eidx = (const int*)d_in[1];
  const int*   src  = eidx;
  const int*   dst  = eidx + E;
  const float* ea   = (const float*)d_in[2];
  const float* W1   = (const float*)d_in[3];
  const float* bl1  = (const float*)d_in[4];
  const float* We1  = (const float*)d_in[5];
  const float* att1 = (const float*)d_in[6];
  const float* b1   = (const float*)d_in[7];
  const float* W2   = (const float*)d_in[8];
  const float* bl2  = (const float*)d_in[9];
  const float* We2  = (const float*)d_in[10];
  const float* att2 = (const float*)d_in[11];
  const float* b2   = (const float*)d_in[12];

  // workspace layout (floats)
  float* ws  = (float*)d_ws;
  float* deg = ws;                         // N
  float* eam = deg + N;                    // N*16  (sum -> mean, in place)
  float* h   = eam + (size_t)N * 16;       // N*64  (layer2 uses first N*32)
  float* acc = h   + (size_t)N * 64;       // N*64
  float* mb  = acc + (size_t)N * 64;       // N
  float* den = mb  + N;                    // N
  float* alp = den + N;                    // E+N
  float* lat = alp + (size_t)(E + N);      // N*64
  float* lg  = lat + (size_t)N * 64;       // N*32

  const int TB = 256;
  const int edge_blocks = cdiv_i((long)E + N, 8);   // 8 waves/block, 1 wave/edge

  // ---- degrees & mean edge attrs (shared by both layers) ----
  fillk<<<cdiv_i(N, TB), TB, 0, stream>>>(deg, 0.0f, N);
  fillk<<<cdiv_i((long)N * 16, TB), TB, 0, stream>>>(eam, 0.0f, N * 16);
  deg_easum_kernel<<<cdiv_i((long)E * 16, TB), TB, 0, stream>>>(dst, ea, deg, eam, E);
  eamean_kernel<<<cdiv_i((long)N * 16, TB), TB, 0, stream>>>(deg, eam, N);

  // ---- layer 1: D=64 ----
  gemm_wmma_f32<<<cdiv_i((long)(N / 16) * 4, 4), 128, 0, stream>>>(x, W1, bl1, h, N, 64);
  fillk<<<cdiv_i(N, TB), TB, 0, stream>>>(mb, -FLT_MAX, N);
  fillk<<<cdiv_i(N, TB), TB, 0, stream>>>(den, 0.0f, N);
  fillk<<<cdiv_i((long)N * 64, TB), TB, 0, stream>>>(acc, 0.0f, N * 64);
  edge_alpha_kernel<<<edge_blocks, TB, 0, stream>>>(h, src, dst, ea, eam, We1, att1,
                                                    alp, mb, E, N, 64);
  edge_aggr_kernel<<<edge_blocks, TB, 0, stream>>>(h, src, dst, alp, mb, den, acc,
                                                   E, N, 64);
  finalize_kernel<<<cdiv_i((long)N * 64, TB), TB, 0, stream>>>(acc, den, b1, lat, N, 64);

  // ---- layer 2: D=32 ----
  gemm_wmma_f32<<<cdiv_i((long)(N / 16) * 2, 4), 128, 0, stream>>>(lat, W2, bl2, h, N, 32);
  fillk<<<cdiv_i(N, TB), TB, 0, stream>>>(mb, -FLT_MAX, N);
  fillk<<<cdiv_i(N, TB), TB, 0, stream>>>(den, 0.0f, N);
  fillk<<<cdiv_i((long)N * 32, TB), TB, 0, stream>>>(acc, 0.0f, N * 32);
  edge_alpha_kernel<<<edge_blocks, TB, 0, stream>>>(h, src, dst, ea, eam, We2, att2,
                                                    alp, mb, E, N, 32);
  edge_aggr_kernel<<<edge_blocks, TB, 0, stream>>>(h, src, dst, alp, mb, den, acc,
                                                   E, N, 32);
  finalize_kernel<<<cdiv_i((long)N * 32, TB), TB, 0, stream>>>(acc, den, b2, lg, N, 32);

  // ---- sample + log_prob ----
  float* out = (float*)d_out;
  sample_kernel<<<cdiv_i(N, 8), TB, 0, stream>>>(lg, out, out + N, N);
}